// GCN_17600775979857
// MI455X (gfx1250) — compile-verified
//
#include <hip/hip_runtime.h>

// GCN 2-layer on MI455X (gfx1250, wave32).
// Strategy: aggregate-then-transform for layer 1 (4-wide atomics),
// transform-then-aggregate for layer 2 (2-wide atomics), with both dense
// transforms fused into one WMMA kernel using v_wmma_f32_16x16x4_f32.

typedef __attribute__((ext_vector_type(2))) float v2f;
typedef __attribute__((ext_vector_type(8))) float v8f;

#define WAVES_PER_BLOCK 8
#define PF_DIST 8192  // edge-stream prefetch distance (elements)

// ---------- degree ----------
__global__ void k_deg_init(float* __restrict__ deg, int n) {
  int i = blockIdx.x * blockDim.x + threadIdx.x;
  if (i < n) deg[i] = 1.0f;  // self-loop contributes 1
}

__global__ void k_deg_count(const int* __restrict__ dst, float* __restrict__ deg, int e) {
  int i = blockIdx.x * blockDim.x + threadIdx.x;
  if (i >= e) return;
  if (i + PF_DIST < e) __builtin_prefetch(&dst[i + PF_DIST], 0, 3);  // global_prefetch_b8
  unsafeAtomicAdd(&deg[dst[i]], 1.0f);
}

// dinv = rsqrt(deg); seed layer-1 aggregation with the self-loop term x[i]*dinv^2
__global__ void k_node_prep(const float* __restrict__ deg, const float4* __restrict__ x4,
                            float* __restrict__ dinv, float4* __restrict__ agg1, int n) {
  int i = blockIdx.x * blockDim.x + threadIdx.x;
  if (i >= n) return;
  float d = rsqrtf(deg[i]);
  dinv[i] = d;
  float s = d * d;
  float4 xi = x4[i];
  agg1[i] = make_float4(xi.x * s, xi.y * s, xi.z * s, xi.w * s);
}

// ---------- layer-1 edge aggregation in 4-dim input space ----------
__global__ void k_l1_edge(const int* __restrict__ src, const int* __restrict__ dst,
                          const float* __restrict__ dinv, const float4* __restrict__ x4,
                          float* __restrict__ agg1, int e) {
  int i = blockIdx.x * blockDim.x + threadIdx.x;
  if (i >= e) return;
  if (i + PF_DIST < e) {
    __builtin_prefetch(&src[i + PF_DIST], 0, 3);
    __builtin_prefetch(&dst[i + PF_DIST], 0, 3);
  }
  int s = src[i], d = dst[i];
  float nrm = dinv[s] * dinv[d];
  float4 xs = x4[s];
  float* p = agg1 + (size_t)d * 4;
  unsafeAtomicAdd(p + 0, xs.x * nrm);
  unsafeAtomicAdd(p + 1, xs.y * nrm);
  unsafeAtomicAdd(p + 2, xs.z * nrm);
  unsafeAtomicAdd(p + 3, xs.w * nrm);
}

// ---------- fused transform: h = relu(agg1@W1 + b1); t = h@W2 ----------
// One wave per 16-node tile. Two chained WMMA stages, h staged via LDS to
// convert D-layout -> A-layout (same-wave LDS ops are in-order).
__global__ __launch_bounds__(WAVES_PER_BLOCK * 32)
void k_transform(const float* __restrict__ agg1, const float* __restrict__ W1,
                 const float* __restrict__ b1, const float* __restrict__ W2,
                 const float* __restrict__ b2, const float* __restrict__ dinv,
                 float* __restrict__ tbuf, float* __restrict__ out, int ntiles) {
  __shared__ float lds[WAVES_PER_BLOCK][16][16];
  const int wave = threadIdx.x >> 5;
  const int lane = threadIdx.x & 31;
  const int tile = blockIdx.x * WAVES_PER_BLOCK + wave;
  if (tile >= ntiles) return;  // wave-uniform exit: EXEC all-1 where WMMA runs
  const int base = tile * 16;
  const int m = lane & 15;     // node row / N column, per layout
  const int half = lane >> 4;  // lane half selects upper K / upper M

  // GEMM1: A = agg1 tile [16x4], B = W1 [4x16]
  // A layout: lane(m,half), VGPR v -> (M=m, K=v+2*half)
  v2f a1;
  {
    const float* row = agg1 + (size_t)(base + m) * 4 + 2 * half;
    a1.x = row[0];
    a1.y = row[1];
  }
  // B layout: lane(m,half), VGPR v -> (K=v+2*half, N=m)
  v2f bm1;
  bm1.x = W1[(2 * half + 0) * 16 + m];
  bm1.y = W1[(2 * half + 1) * 16 + m];
  v8f c1 = {};
  c1 = __builtin_amdgcn_wmma_f32_16x16x4_f32(false, a1, false, bm1, (short)0, c1,
                                             false, false);

  // bias + relu; stage h into LDS. D layout: VGPR r -> (M=r+8*half, N=m)
  const float bias = b1[m];
  #pragma unroll
  for (int r = 0; r < 8; ++r) {
    float h = c1[r] + bias;
    h = h > 0.f ? h : 0.f;
    lds[wave][r + 8 * half][m] = h;
  }

  // GEMM2: t = h[16x16] @ W2pad[16x16] (W2 zero-padded to 16 cols), K in 4 chunks.
  // Branchless padded-B: clamp column index (W2 has 2 cols), zero value by mask.
  const float cmask = (m < 2) ? 1.f : 0.f;
  const int cidx = m & 1;
  v8f c2 = {};
  #pragma unroll
  for (int j = 0; j < 4; ++j) {
    v2f a2, bm2;
    a2.x = lds[wave][m][4 * j + 2 * half + 0];
    a2.y = lds[wave][m][4 * j + 2 * half + 1];
    bm2.x = W2[(4 * j + 2 * half + 0) * 2 + cidx] * cmask;
    bm2.y = W2[(4 * j + 2 * half + 1) * 2 + cidx] * cmask;
    c2 = __builtin_amdgcn_wmma_f32_16x16x4_f32(false, a2, false, bm2, (short)0, c2,
                                               false, false);
  }

  // stage t tile, then write t and initialize out with bias + self-loop term
  #pragma unroll
  for (int r = 0; r < 8; ++r) lds[wave][r + 8 * half][m] = c2[r];

  {
    int m2 = lane >> 1, nn = lane & 1;  // 16 nodes x 2 cols = 32 lanes
    int node = base + m2;
    float tv = lds[wave][m2][nn];
    float dv = dinv[node];
    tbuf[(size_t)node * 2 + nn] = tv;
    out[(size_t)node * 2 + nn] = b2[nn] + tv * dv * dv;
  }
}

// scalar tail for n % 16 != 0 (not taken for n=100000, kept for safety)
__global__ void k_transform_tail(const float* __restrict__ agg1, const float* __restrict__ W1,
                                 const float* __restrict__ b1, const float* __restrict__ W2,
                                 const float* __restrict__ b2, const float* __restrict__ dinv,
                                 float* __restrict__ tbuf, float* __restrict__ out,
                                 int start, int n) {
  int i = start + blockIdx.x * blockDim.x + threadIdx.x;
  if (i >= n) return;
  float h[16];
  for (int o = 0; o < 16; ++o) {
    float acc = b1[o];
    for (int k = 0; k < 4; ++k) acc += agg1[(size_t)i * 4 + k] * W1[k * 16 + o];
    h[o] = acc > 0.f ? acc : 0.f;
  }
  float dv = dinv[i];
  for (int o = 0; o < 2; ++o) {
    float acc = 0.f;
    for (int k = 0; k < 16; ++k) acc += h[k] * W2[k * 2 + o];
    tbuf[(size_t)i * 2 + o] = acc;
    out[(size_t)i * 2 + o] = b2[o] + acc * dv * dv;
  }
}

// ---------- layer-2 edge aggregation in 2-dim output space ----------
__global__ void k_l2_edge(const int* __restrict__ src, const int* __restrict__ dst,
                          const float* __restrict__ dinv, const float2* __restrict__ tbuf,
                          float* __restrict__ out, int e) {
  int i = blockIdx.x * blockDim.x + threadIdx.x;
  if (i >= e) return;
  if (i + PF_DIST < e) {
    __builtin_prefetch(&src[i + PF_DIST], 0, 3);
    __builtin_prefetch(&dst[i + PF_DIST], 0, 3);
  }
  int s = src[i], d = dst[i];
  float nrm = dinv[s] * dinv[d];
  float2 ts = tbuf[s];
  unsafeAtomicAdd(&out[(size_t)d * 2 + 0], ts.x * nrm);
  unsafeAtomicAdd(&out[(size_t)d * 2 + 1], ts.y * nrm);
}

extern "C" void kernel_launch(void* const* d_in, const int* in_sizes, int n_in,
                              void* d_out, int out_size, void* d_ws, size_t ws_size,
                              hipStream_t stream) {
  const float* x  = (const float*)d_in[0];
  const int*   ei = (const int*)d_in[1];
  const float* W1 = (const float*)d_in[2];
  const float* b1 = (const float*)d_in[3];
  const float* W2 = (const float*)d_in[4];
  const float* b2 = (const float*)d_in[5];
  float* out = (float*)d_out;

  const int n = in_sizes[0] / 4;   // 100000
  const int e = in_sizes[1] / 2;   // 3200000
  const int* src = ei;
  const int* dst = ei + e;

  // workspace carve-out: deg[n] | dinv[n] | agg1[n*4] | t[n*2]  (= 32B/node)
  char* ws = (char*)d_ws;
  float* deg  = (float*)(ws);
  float* dinv = (float*)(ws + (size_t)n * 4);
  float* agg1 = (float*)(ws + (size_t)n * 8);
  float* tbuf = (float*)(ws + (size_t)n * 24);

  const int T = 256;
  const int gn = (n + T - 1) / T;
  const int ge = (e + T - 1) / T;

  k_deg_init<<<gn, T, 0, stream>>>(deg, n);
  k_deg_count<<<ge, T, 0, stream>>>(dst, deg, e);
  k_node_prep<<<gn, T, 0, stream>>>(deg, (const float4*)x, dinv, (float4*)agg1, n);
  k_l1_edge<<<ge, T, 0, stream>>>(src, dst, dinv, (const float4*)x, agg1, e);

  const int ntiles = n / 16;
  if (ntiles > 0) {
    const int gt = (ntiles + WAVES_PER_BLOCK - 1) / WAVES_PER_BLOCK;
    k_transform<<<gt, WAVES_PER_BLOCK * 32, 0, stream>>>(agg1, W1, b1, W2, b2, dinv,
                                                         tbuf, out, ntiles);
  }
  const int rem = n - ntiles * 16;
  if (rem > 0) {
    k_transform_tail<<<(rem + T - 1) / T, T, 0, stream>>>(agg1, W1, b1, W2, b2, dinv,
                                                          tbuf, out, ntiles * 16, n);
  }

  k_l2_edge<<<ge, T, 0, stream>>>(src, dst, dinv, (const float2*)tbuf, out, e);
}